// OL_Operator_18408229830711
// MI455X (gfx1250) — compile-verified
//
#include <hip/hip_runtime.h>
#include <stdint.h>

typedef uint32_t u32;
typedef __attribute__((ext_vector_type(4))) u32 u32x4;
typedef __attribute__((ext_vector_type(8))) u32 u32x8;

#define OL_THRESH 0.997f
#define OL_NLV    7.0f

// state block in d_ws (u32 words)
#define ST_ZEROS  0
#define ST_PREFIX 1
#define ST_RANK   2
#define ST_TH     3
#define ST_STEP   4

#define H1_BINS 2048   // bits [30:20] of |x| bit pattern
#define H2_BINS 1024   // refine 10 bits at a time

// ---------------- init: zero histograms (head of d_out) + state (d_ws) ----
__global__ void k_init(u32* __restrict__ hist, u32* __restrict__ state) {
  int i = blockIdx.x * blockDim.x + threadIdx.x;
  if (i < 4096) hist[i] = 0u;
  if (i < 16)   state[i] = 0u;
}

// ---------------- pass 1: histogram of key>>20, count exact zeros ---------
__global__ void k_hist_hi(const float* __restrict__ x, u32* __restrict__ hist,
                          u32* __restrict__ state, int n) {
  __shared__ u32 lh[H1_BINS];
  for (int i = threadIdx.x; i < H1_BINS; i += blockDim.x) lh[i] = 0u;
  __syncthreads();
  int gtid = blockIdx.x * blockDim.x + threadIdx.x;
  int gs   = gridDim.x * blockDim.x;
  int n4   = n >> 2;
  const float4* x4 = (const float4*)x;
  u32 zc = 0u;
  for (int i = gtid; i < n4; i += gs) {
    float4 v = x4[i];
    u32 k0 = __float_as_uint(v.x) & 0x7fffffffu;
    u32 k1 = __float_as_uint(v.y) & 0x7fffffffu;
    u32 k2 = __float_as_uint(v.z) & 0x7fffffffu;
    u32 k3 = __float_as_uint(v.w) & 0x7fffffffu;
    atomicAdd(&lh[k0 >> 20], 1u);
    atomicAdd(&lh[k1 >> 20], 1u);
    atomicAdd(&lh[k2 >> 20], 1u);
    atomicAdd(&lh[k3 >> 20], 1u);
    zc += (k0 == 0u) + (k1 == 0u) + (k2 == 0u) + (k3 == 0u);
  }
  for (int i = (n4 << 2) + gtid; i < n; i += gs) {
    u32 k = __float_as_uint(x[i]) & 0x7fffffffu;
    atomicAdd(&lh[k >> 20], 1u);
    zc += (k == 0u);
  }
  if (zc) atomicAdd(&state[ST_ZEROS], zc);
  __syncthreads();
  for (int i = threadIdx.x; i < H1_BINS; i += blockDim.x)
    if (lh[i]) atomicAdd(&hist[i], lh[i]);
}

// ---------------- refine passes: histogram 10 bits under a matched prefix -
__global__ void k_hist_sub(const float* __restrict__ x, u32* __restrict__ hist,
                           const u32* __restrict__ state, int n,
                           int matchShift, int binShift) {
  __shared__ u32 lh[H2_BINS];
  for (int i = threadIdx.x; i < H2_BINS; i += blockDim.x) lh[i] = 0u;
  __syncthreads();
  u32 prefix = state[ST_PREFIX];   // uniform
  int gtid = blockIdx.x * blockDim.x + threadIdx.x;
  int gs   = gridDim.x * blockDim.x;
  int n4   = n >> 2;
  const float4* x4 = (const float4*)x;
  for (int i = gtid; i < n4; i += gs) {
    float4 v = x4[i];
    u32 k0 = __float_as_uint(v.x) & 0x7fffffffu;
    u32 k1 = __float_as_uint(v.y) & 0x7fffffffu;
    u32 k2 = __float_as_uint(v.z) & 0x7fffffffu;
    u32 k3 = __float_as_uint(v.w) & 0x7fffffffu;
    if ((k0 >> matchShift) == prefix) atomicAdd(&lh[(k0 >> binShift) & 1023u], 1u);
    if ((k1 >> matchShift) == prefix) atomicAdd(&lh[(k1 >> binShift) & 1023u], 1u);
    if ((k2 >> matchShift) == prefix) atomicAdd(&lh[(k2 >> binShift) & 1023u], 1u);
    if ((k3 >> matchShift) == prefix) atomicAdd(&lh[(k3 >> binShift) & 1023u], 1u);
  }
  for (int i = (n4 << 2) + gtid; i < n; i += gs) {
    u32 k = __float_as_uint(x[i]) & 0x7fffffffu;
    if ((k >> matchShift) == prefix) atomicAdd(&lh[(k >> binShift) & 1023u], 1u);
  }
  __syncthreads();
  for (int i = threadIdx.x; i < H2_BINS; i += blockDim.x)
    if (lh[i]) atomicAdd(&hist[i], lh[i]);
}

// ---------------- select the bin holding the target rank ------------------
__global__ void k_select(const u32* __restrict__ hist, u32* __restrict__ state,
                         int nbins, int pass, int n) {
  __shared__ u32 psum[256];
  int tid = threadIdx.x;
  int per = nbins >> 8;
  u32 s = 0u;
  for (int j = 0; j < per; ++j) s += hist[tid * per + j];
  psum[tid] = s;
  __syncthreads();
  if (tid == 0) {
    u32 rank;
    if (pass == 0) {
      u32 zeros = state[ST_ZEROS];
      u32 nnz = (u32)n - zeros;
      // reproduce: floor(0.997f * float(nnz)).astype(int32) + (n - nnz)
      float fidx = floorf(OL_THRESH * (float)nnz);
      long long idx = (long long)fidx + (long long)zeros;
      if (idx > (long long)n - 1) idx = (long long)n - 1;
      if (idx < 0) idx = 0;
      rank = (u32)idx;
    } else {
      rank = state[ST_RANK];
    }
    int b = -1; u32 within = 0u; u32 c = 0u;
    for (int i = 0; i < 256 && b < 0; ++i) {
      u32 p = psum[i];
      if (c + p > rank) {
        u32 cc = c;
        for (int j = 0; j < per; ++j) {
          u32 h = hist[i * per + j];
          if (cc + h > rank) { b = i * per + j; within = rank - cc; break; }
          cc += h;
        }
      }
      c += p;
    }
    if (b < 0) { b = nbins - 1; within = 0u; }
    u32 prefix = (pass == 0) ? (u32)b : ((state[ST_PREFIX] << 10) | (u32)b);
    state[ST_PREFIX] = prefix;
    state[ST_RANK]   = within;
    if (pass == 2) {
      state[ST_TH] = prefix;                       // 31-bit key == f32 bits of th
      float th = __uint_as_float(prefix);
      state[ST_STEP] = __float_as_uint(th / OL_NLV);
    }
  }
}

// ---------------- TDM-pipelined elementwise quantize ----------------------
#define QTILE  1024           // f32 elements per TDM tile (4 KB)
#define QV4    (QTILE / 4)
#define QWAVES 4              // 128-thread blocks -> 4 waves, 32 KB LDS

// Build a 1-D TDM descriptor and issue tensor_load_to_lds (2-group form).
__device__ __forceinline__ void tdm_load(const float* gsrc, u32 lds_off, u32 elems) {
  unsigned long long ga = (unsigned long long)(uintptr_t)gsrc;
  u32x4 g0;
  g0[0] = 1u;                                            // count=1, valid descriptor
  g0[1] = lds_off;                                       // LDS byte address
  g0[2] = (u32)ga;                                       // global_addr[31:0]
  g0[3] = ((u32)(ga >> 32) & 0x01ffffffu) | 0x80000000u; // global_addr[56:32] | type=2
  u32x8 g1;
  g1[0] = 0x00020000u;                 // data_size = 2 (4 bytes)
  g1[1] = (elems & 0xffffu) << 16;     // tensor_dim0[15:0]  (OOB reads -> 0)
  g1[2] = (elems >> 16) | (1u << 16);  // tensor_dim0[31:16] | tensor_dim1 = 1
  g1[3] = ((u32)QTILE) << 16;          // tile_dim0 = QTILE
  g1[4] = 0u;                          // tile_dim1 = tile_dim2 = 0 (1-D tile)
  g1[5] = elems;                       // tensor_dim0_stride (unused, 1 row)
  g1[6] = 0u;
  g1[7] = 0u;
  asm volatile("tensor_load_to_lds %0, %1" :: "s"(g0), "s"(g1) : "memory");
}

__device__ __forceinline__ float ol_quant(float v, float th, float step) {
  // round-half-even matches jnp.round; th==0 -> pass-through (matches where(th>0,...))
  float q = __builtin_rintf(v / step) * step;
  return (th > 0.0f && __builtin_fabsf(v) <= th) ? q : v;
}

__global__ void __launch_bounds__(128)
k_quantize(const float* __restrict__ x, float* __restrict__ out,
           const u32* __restrict__ state, int n) {
  __shared__ float4 tiles[QWAVES * 2 * QV4];   // double buffer per wave
  float th   = __uint_as_float(state[ST_TH]);
  float step = __uint_as_float(state[ST_STEP]);
  int wid  = threadIdx.x >> 5;
  int lane = threadIdx.x & 31;
  int gw = blockIdx.x * QWAVES + wid;
  int nw = gridDim.x * QWAVES;
  int ntiles = (n + QTILE - 1) / QTILE;
  u32 lds_base = (u32)(uintptr_t)(&tiles[0]) + (u32)wid * 2u * (QTILE * 4u);

  int t = gw;
  if (t < ntiles) {
    int base = t * QTILE;
    int rem  = n - base; if (rem > QTILE) rem = QTILE;
    tdm_load(x + base, lds_base, (u32)rem);
  }
  int buf = 0;
  for (; t < ntiles; t += nw) {
    int tn = t + nw;
    if (tn < ntiles) {
      int bn = tn * QTILE;
      int rn = n - bn; if (rn > QTILE) rn = QTILE;
      tdm_load(x + bn, lds_base + (u32)(buf ^ 1) * (QTILE * 4u), (u32)rn);
      asm volatile("s_wait_tensorcnt 0x1" ::: "memory");  // current tile resident
    } else {
      asm volatile("s_wait_tensorcnt 0x0" ::: "memory");
    }
    const float4* lt = tiles + (wid * 2 + buf) * QV4;
    int base = t * QTILE;
    if (base + QTILE <= n) {
      float4* o4 = (float4*)(out + base);
#pragma unroll
      for (int j = 0; j < QV4 / 32; ++j) {
        int e = lane + 32 * j;
        float4 v = lt[e];                       // ds_load_b128
        float4 r;
        r.x = ol_quant(v.x, th, step);
        r.y = ol_quant(v.y, th, step);
        r.z = ol_quant(v.z, th, step);
        r.w = ol_quant(v.w, th, step);
        o4[e] = r;                              // global_store_b128
      }
    } else {
      const float* lf = (const float*)lt;
      int rem = n - base;
      for (int e = lane; e < rem; e += 32)
        out[base + e] = ol_quant(lf[e], th, step);
    }
    buf ^= 1;
  }
}

// -------------------------------------------------------------------------
extern "C" void kernel_launch(void* const* d_in, const int* in_sizes, int n_in,
                              void* d_out, int out_size, void* d_ws, size_t ws_size,
                              hipStream_t stream) {
  (void)n_in; (void)out_size; (void)ws_size;
  const float* x = (const float*)d_in[0];
  float* out = (float*)d_out;
  int n = in_sizes[0];
  if (n <= 0) return;

  // Histograms live at the head of d_out (fully overwritten by k_quantize);
  // only the 64-byte state block uses d_ws.
  u32* hist  = (u32*)d_out;
  u32* state = (u32*)d_ws;

  hipLaunchKernelGGL(k_init,     dim3(16),   dim3(256), 0, stream, hist, state);
  hipLaunchKernelGGL(k_hist_hi,  dim3(1024), dim3(256), 0, stream, x, hist, state, n);
  hipLaunchKernelGGL(k_select,   dim3(1),    dim3(256), 0, stream, hist, state, H1_BINS, 0, n);
  hipLaunchKernelGGL(k_hist_sub, dim3(1024), dim3(256), 0, stream, x, hist + 2048, state, n, 20, 10);
  hipLaunchKernelGGL(k_select,   dim3(1),    dim3(256), 0, stream, hist + 2048, state, H2_BINS, 1, n);
  hipLaunchKernelGGL(k_hist_sub, dim3(1024), dim3(256), 0, stream, x, hist + 3072, state, n, 10, 0);
  hipLaunchKernelGGL(k_select,   dim3(1),    dim3(256), 0, stream, hist + 3072, state, H2_BINS, 2, n);
  hipLaunchKernelGGL(k_quantize, dim3(1024), dim3(128), 0, stream, x, out, state, n);
}